// BI_LSTM_CRF_MODEL_89421219103628
// MI455X (gfx1250) — compile-verified
//
#include <hip/hip_runtime.h>
#include <hip/hip_bf16.h>

// ---------------- problem constants ----------------
#define LSEQ   4096
#define EDIM   256
#define H2DIM  256
#define G4     1024          // 4 * H2
#define HDIM   512
#define TTAGS  32
#define TSTART 30
#define TSTOP  31
#define NEGINF (-10000.0f)
#define JLDS   34            // W_hh^T rows pinned in LDS (34*1024*4B = 136KB)

typedef __attribute__((ext_vector_type(2))) float v2f;
typedef __attribute__((ext_vector_type(8))) float v8f;

// ---------------------------------------------------------------------------
// Kernel 1: embedding gather  x[l, :] = emb[sentence[l], :]
// ---------------------------------------------------------------------------
__global__ void k_gather_x(const int* __restrict__ sent,
                           const float* __restrict__ emb,
                           float* __restrict__ x) {
    int l = blockIdx.x;
    int e = threadIdx.x;              // blockDim = 256 = EDIM
    x[(size_t)l * EDIM + e] = emb[(size_t)sent[l] * EDIM + e];
}

// ---------------------------------------------------------------------------
// Kernel 2: transpose W_hh [1024,256] -> WT [256,1024]
// (makes the per-step LSTM GEMV loads coalesced: lane g reads WT[j*1024+g])
// ---------------------------------------------------------------------------
__global__ void k_transpose_whh(const float* __restrict__ W,
                                float* __restrict__ WT) {
    int idx = blockIdx.x * 256 + threadIdx.x;   // total G4*H2 = 262144
    int r = idx >> 8;                           // 0..1023 (gate row)
    int c = idx & 255;                          // 0..255  (h index)
    WT[(size_t)c * G4 + r] = W[idx];
}

// ---------------------------------------------------------------------------
// Kernel 3: input projections via fp32 WMMA (V_WMMA_F32_16X16X4_F32)
//   xW = x[L,256] @ W_ih.T[256,1024]    (one wave per 16x16 output tile)
// A frag (16x4 f32): v[0]=K(+0), v[1]=K(+1); lanes 16..31 use K+2. M = lane%16
// B frag (4x16 f32): same pattern, N = lane%16 (B col n = W_ih row n)
// D (16x16 f32):     VGPR r -> row r + 8*(lane>=16), col = lane%16
// ---------------------------------------------------------------------------
__global__ void k_gemm_xw(const float* __restrict__ x,
                          const float* __restrict__ Wf,
                          const float* __restrict__ Wb,
                          float* __restrict__ outf,
                          float* __restrict__ outb) {
    const float* W   = blockIdx.y ? Wb   : Wf;
    float*       out = blockIdx.y ? outb : outf;

    int wave = threadIdx.x >> 5;                 // 8 waves / block
    int lane = threadIdx.x & 31;
    int tile = blockIdx.x * 8 + wave;            // 0 .. 16383
    const int tiles_per_row = G4 / 16;           // 64
    int row0 = (tile / tiles_per_row) * 16;
    int col0 = (tile % tiles_per_row) * 16;

    int lm = lane & 15;
    int hi = lane >> 4;                          // 0 | 1

    const float* arow = x + (size_t)(row0 + lm) * EDIM;
    const float* brow = W + (size_t)(col0 + lm) * EDIM;

    v8f acc = {};
#pragma unroll 4
    for (int k = 0; k < EDIM; k += 4) {
        v2f a = *(const v2f*)(arow + k + 2 * hi);
        v2f b = *(const v2f*)(brow + k + 2 * hi);
        acc = __builtin_amdgcn_wmma_f32_16x16x4_f32(
            /*neg_a=*/false, a, /*neg_b=*/false, b,
            /*c_mod=*/(short)0, acc, /*reuse_a=*/false, /*reuse_b=*/false);
    }

    float* obase = out + (size_t)row0 * G4 + col0 + lm;
#pragma unroll
    for (int r = 0; r < 8; ++r)
        obase[(size_t)(r + 8 * hi) * G4] = acc[r];
}

// ---------------------------------------------------------------------------
// Kernel 4: bidirectional LSTM recurrence. One 1024-thread block per direction.
// Thread g owns gate row g: acc = xW[t][g] + b[g] + sum_j WT[j][g] * h[j].
// Gate order (torch): i, f, g, o in chunks of 256.
//  * rows j < JLDS of WT are pinned in LDS for the whole scan (L2 traffic cut)
//  * next step's xW row is prefetched into LDS via GLOBAL_LOAD_ASYNC_TO_LDS_B32
//    (each lane prefetches exactly the word it reads -> per-wave asynccnt wait)
// ---------------------------------------------------------------------------
__global__ void k_lstm(const float* __restrict__ xWf,
                       const float* __restrict__ xWb,
                       const float* __restrict__ WTf,
                       const float* __restrict__ WTb,
                       const float* __restrict__ b_f,
                       const float* __restrict__ b_b,
                       const float* __restrict__ h0,
                       const float* __restrict__ c0,
                       float* __restrict__ hf,
                       float* __restrict__ hb) {
    const int dir = blockIdx.x;                  // 0 = fwd, 1 = bwd
    const float* xW   = dir ? xWb : xWf;
    const float* WT   = dir ? WTb : WTf;
    const float* bias = dir ? b_b : b_f;
    float*       hout = dir ? hb  : hf;

    __shared__ float h_s[H2DIM];
    __shared__ float c_s[H2DIM];
    __shared__ float gates_s[G4];
    __shared__ float xw_s[2][G4];                // double-buffered xW[t] row
    __shared__ float w_s[JLDS * G4];             // LDS-resident W slice (136KB)

    const int g = threadIdx.x;                   // 0..1023
    if (g < H2DIM) {
        h_s[g] = h0[dir * H2DIM + g];
        c_s[g] = c0[dir * H2DIM + g];
    }
    // pin first JLDS rows of WT in LDS (coalesced, done once)
#pragma unroll 2
    for (int j = 0; j < JLDS; ++j)
        w_s[j * G4 + g] = WT[(size_t)j * G4 + g];

    const float bg = bias[g];
    const float* wcol = WT + g;                  // stride G4 over j

    // initial async prefetch of the first xW row into buffer 0
    {
        int t0 = dir ? (LSEQ - 1) : 0;
        unsigned lds0 = (unsigned)(uintptr_t)&xw_s[0][g];
        unsigned long long src0 =
            (unsigned long long)(uintptr_t)(xW + (size_t)t0 * G4 + g);
        asm volatile("global_load_async_to_lds_b32 %0, %1, off"
                     :: "v"(lds0), "v"(src0) : "memory");
    }
    __syncthreads();

    for (int s = 0; s < LSEQ; ++s) {
        const int t   = dir ? (LSEQ - 1 - s) : s;
        const int cur = s & 1;
        const int nxt = cur ^ 1;

        // wait for this step's prefetched xW row (per-wave ASYNCcnt)
        asm volatile("s_wait_asynccnt 0x0" ::: "memory");
        float acc = xw_s[cur][g] + bg;

        // issue prefetch for next step's row into the other buffer
        {
            int sn = (s + 1 < LSEQ) ? (s + 1) : s;
            int tn = dir ? (LSEQ - 1 - sn) : sn;
            unsigned ldsn = (unsigned)(uintptr_t)&xw_s[nxt][g];
            unsigned long long srcn =
                (unsigned long long)(uintptr_t)(xW + (size_t)tn * G4 + g);
            asm volatile("global_load_async_to_lds_b32 %0, %1, off"
                         :: "v"(ldsn), "v"(srcn) : "memory");
        }

        // GEMV: LDS-resident slice first, then L2-resident remainder
#pragma unroll 2
        for (int j = 0; j < JLDS; ++j)
            acc = fmaf(w_s[j * G4 + g], h_s[j], acc);
#pragma unroll 8
        for (int j = JLDS; j < H2DIM; ++j)
            acc = fmaf(wcol[(size_t)j * G4], h_s[j], acc);

        gates_s[g] = acc;
        __syncthreads();

        if (g < H2DIM) {
            float ig = 1.0f / (1.0f + __expf(-gates_s[g]));
            float fg = 1.0f / (1.0f + __expf(-gates_s[H2DIM + g]));
            float gg = tanhf(gates_s[2 * H2DIM + g]);
            float og = 1.0f / (1.0f + __expf(-gates_s[3 * H2DIM + g]));
            float c  = fg * c_s[g] + ig * gg;
            float h  = og * tanhf(c);
            c_s[g] = c;
            h_s[g] = h;
            hout[(size_t)t * H2DIM + g] = h;
        }
        __syncthreads();
    }
}

// ---------------------------------------------------------------------------
// Kernel 5: feats = [hf|hb][L,512] @ W_out.T[512,32] + b_out, via fp32 WMMA.
// 256 row-tiles x 2 col-tiles = 512 waves.
// ---------------------------------------------------------------------------
__global__ void k_gemm_feats(const float* __restrict__ hf,
                             const float* __restrict__ hb,
                             const float* __restrict__ Wout,
                             const float* __restrict__ bout,
                             float* __restrict__ feats) {
    int wave = threadIdx.x >> 5;
    int lane = threadIdx.x & 31;
    int tile = blockIdx.x * 8 + wave;            // 0 .. 511
    int row0 = (tile >> 1) * 16;
    int col0 = (tile & 1) * 16;

    int lm = lane & 15;
    int hi = lane >> 4;

    const float* arow_f = hf + (size_t)(row0 + lm) * H2DIM;
    const float* arow_b = hb + (size_t)(row0 + lm) * H2DIM;
    const float* brow   = Wout + (size_t)(col0 + lm) * HDIM;

    v8f acc = {};
#pragma unroll 4
    for (int k = 0; k < HDIM; k += 4) {
        const float* ar = (k < H2DIM) ? (arow_f + k) : (arow_b + (k - H2DIM));
        v2f a = *(const v2f*)(ar + 2 * hi);
        v2f b = *(const v2f*)(brow + k + 2 * hi);
        acc = __builtin_amdgcn_wmma_f32_16x16x4_f32(
            false, a, false, b, (short)0, acc, false, false);
    }

    float bn = bout[col0 + lm];
#pragma unroll
    for (int r = 0; r < 8; ++r)
        feats[(size_t)(row0 + r + 8 * hi) * TTAGS + col0 + lm] = acc[r] + bn;
}

// ---------------------------------------------------------------------------
// Kernel 6: CRF forward scan ((logsumexp,+) semiring) + gold path score.
// One wave32 block; lane j owns tag j. Barriers in a 1-wave WG are S_NOPs.
// ---------------------------------------------------------------------------
__global__ void k_crf(const float* __restrict__ feats,
                      const float* __restrict__ trans,
                      const int* __restrict__ tags,
                      float* __restrict__ out) {
    __shared__ float prev_s[TTAGS];
    __shared__ float tr_s[TTAGS * TTAGS];

    const int j = threadIdx.x;                   // 0..31
    for (int i = 0; i < TTAGS; ++i)
        tr_s[i * TTAGS + j] = trans[i * TTAGS + j];

    float prev = (j == TSTART) ? 0.0f : NEGINF;
    prev_s[j] = prev;
    __syncthreads();

    for (int t = 0; t < LSEQ; ++t) {
        float v[TTAGS];
        float m = -3.4e38f;
#pragma unroll
        for (int i = 0; i < TTAGS; ++i) {
            v[i] = prev_s[i] + tr_s[i * TTAGS + j];
            m = fmaxf(m, v[i]);
        }
        float ssum = 0.0f;
#pragma unroll
        for (int i = 0; i < TTAGS; ++i)
            ssum += __expf(v[i] - m);
        prev = m + __logf(ssum) + feats[(size_t)t * TTAGS + j];
        __syncthreads();
        prev_s[j] = prev;
        __syncthreads();
    }

    // forward_score = logsumexp_j(prev[j] + trans[j, STOP])
    float fv = prev + tr_s[j * TTAGS + TSTOP];
    float m = fv;
    for (int off = 16; off > 0; off >>= 1) m = fmaxf(m, __shfl_down(m, off));
    m = __shfl(m, 0);
    float e = __expf(fv - m);
    for (int off = 16; off > 0; off >>= 1) e += __shfl_down(e, off);
    e = __shfl(e, 0);
    float fwd = m + __logf(e);

    // gold path score (strided over lanes)
    float ps = 0.0f;
    for (int l = j; l < LSEQ; l += TTAGS) {
        int tg = tags[l];
        int tp = (l == 0) ? TSTART : tags[l - 1];
        ps += feats[(size_t)l * TTAGS + tg] + tr_s[tp * TTAGS + tg];
    }
    for (int off = 16; off > 0; off >>= 1) ps += __shfl_down(ps, off);

    if (j == 0) {
        ps += tr_s[tags[LSEQ - 1] * TTAGS + TSTOP];
        out[0] = fwd - ps;
    }
}

// ---------------------------------------------------------------------------
// Host-side launcher
// ---------------------------------------------------------------------------
extern "C" void kernel_launch(void* const* d_in, const int* in_sizes, int n_in,
                              void* d_out, int out_size, void* d_ws, size_t ws_size,
                              hipStream_t stream) {
    const int*   sentence = (const int*)d_in[0];
    const int*   tags     = (const int*)d_in[1];
    const float* emb      = (const float*)d_in[2];
    const float* W_ih_f   = (const float*)d_in[3];
    const float* W_hh_f   = (const float*)d_in[4];
    const float* b_f      = (const float*)d_in[5];
    const float* W_ih_b   = (const float*)d_in[6];
    const float* W_hh_b   = (const float*)d_in[7];
    const float* b_b      = (const float*)d_in[8];
    const float* W_out    = (const float*)d_in[9];
    const float* b_out    = (const float*)d_in[10];
    const float* trans    = (const float*)d_in[11];
    const float* h0       = (const float*)d_in[12];
    const float* c0       = (const float*)d_in[13];
    float*       out      = (float*)d_out;

    // workspace carve-up (floats)
    float* ws    = (float*)d_ws;
    float* x     = ws;                            // L*E  = 1,048,576
    float* xWf   = x    + (size_t)LSEQ * EDIM;    // L*G4 = 4,194,304
    float* xWb   = xWf  + (size_t)LSEQ * G4;      // L*G4
    float* WTf   = xWb  + (size_t)LSEQ * G4;      // E*G4 =   262,144
    float* WTb   = WTf  + (size_t)EDIM * G4;
    float* hfbuf = WTb  + (size_t)EDIM * G4;      // L*H2 = 1,048,576
    float* hbbuf = hfbuf + (size_t)LSEQ * H2DIM;
    float* feats = hbbuf + (size_t)LSEQ * H2DIM;  // L*T  =   131,072

    // 1. embedding gather
    k_gather_x<<<LSEQ, EDIM, 0, stream>>>(sentence, emb, x);

    // 2. transpose recurrent weights for coalesced GEMV
    k_transpose_whh<<<(G4 * H2DIM) / 256, 256, 0, stream>>>(W_hh_f, WTf);
    k_transpose_whh<<<(G4 * H2DIM) / 256, 256, 0, stream>>>(W_hh_b, WTb);

    // 3. input projections (WMMA f32): 16384 tiles per direction, 8 waves/block
    dim3 g1((LSEQ / 16) * (G4 / 16) / 8, 2);
    k_gemm_xw<<<g1, 256, 0, stream>>>(x, W_ih_f, W_ih_b, xWf, xWb);

    // 4. sequential biLSTM (one persistent workgroup per direction)
    k_lstm<<<2, G4, 0, stream>>>(xWf, xWb, WTf, WTb, b_f, b_b, h0, c0,
                                 hfbuf, hbbuf);

    // 5. output projection (WMMA f32): 512 tiles, 8 waves/block
    k_gemm_feats<<<(LSEQ / 16) * 2 / 8, 256, 0, stream>>>(hfbuf, hbbuf,
                                                          W_out, b_out, feats);

    // 6. CRF forward scan + path score -> scalar
    k_crf<<<1, TTAGS, 0, stream>>>(feats, trans, tags, out);
}